// SceneFlowEstimatorInit_35648228557119
// MI455X (gfx1250) — compile-verified
//
#include <hip/hip_runtime.h>
#include <cmath>
#include <stdint.h>

// ---------------- problem constants ----------------
#define B_    2
#define N_    4096
#define C_    64
#define D_    128
#define K_    32
#define KPC_  16
#define WN_   8
#define SLOPE_ 0.1f
#define EPS_   1e-5f
#define CNT_BN_ (B_*N_)          // 8192   (BN over axes (0,2))
#define CNT_WN_ (B_*N_*KPC_)     // 131072 (BN over axes (0,2,3))

typedef __attribute__((ext_vector_type(2))) float v2f;
typedef __attribute__((ext_vector_type(8))) float v8f;

__device__ __forceinline__ float lrelu_(float x) { return x >= 0.f ? x : SLOPE_ * x; }

// =====================================================================
// 1) Exact KNN: one wave per query point. Each lane keeps a sorted
//    top-32 list of its 128 candidates in registers; lists are merged
//    through LDS with a shuffle-based lexicographic argmin (wave32).
// =====================================================================
__global__ __launch_bounds__(128) void knn_kernel(const float* __restrict__ xyz,
                                                  int* __restrict__ idx_out) {
  __shared__ float sx[128], sy[128], sz[128];
  __shared__ float md[4][32 * 32];
  __shared__ int   mi[4][32 * 32];

  const int lane = threadIdx.x & 31;
  const int wv   = threadIdx.x >> 5;
  const int col  = blockIdx.x * 4 + wv;     // flat (b, n)
  const int b    = col / N_;
  const int n    = col % N_;
  const size_t xb = (size_t)b * 3 * N_;

  const float qx = xyz[xb + n], qy = xyz[xb + N_ + n], qz = xyz[xb + 2 * N_ + n];

  float bd[32]; int bi[32];
#pragma unroll
  for (int j = 0; j < 32; ++j) { bd[j] = 3.402823e38f; bi[j] = 0x7fffffff; }

  for (int c0 = 0; c0 < N_; c0 += 128) {
    __syncthreads();
    {
      const int m = c0 + threadIdx.x;
      sx[threadIdx.x] = xyz[xb + m];
      sy[threadIdx.x] = xyz[xb + N_ + m];
      sz[threadIdx.x] = xyz[xb + 2 * N_ + m];
      if (c0 + 128 < N_) {   // hint next chunk toward L2 (global_prefetch_b8)
        __builtin_prefetch(&xyz[xb + m + 128], 0, 1);
      }
    }
    __syncthreads();
#pragma unroll
    for (int s = 0; s < 4; ++s) {
      const int t = lane + 32 * s;
      const float dx = sx[t] - qx, dy = sy[t] - qy, dz = sz[t] - qz;
      const float d = dx * dx + dy * dy + dz * dz;
      const int m = c0 + t;
      if (d < bd[31] || (d == bd[31] && m < bi[31])) {
        float nd = d; int ni = m;
#pragma unroll
        for (int j = 0; j < 32; ++j) {
          const bool lt = (nd < bd[j]) || (nd == bd[j] && ni < bi[j]);
          const float td = bd[j]; const int ti = bi[j];
          if (lt) { bd[j] = nd; bi[j] = ni; nd = td; ni = ti; }
        }
      }
    }
  }
  // dump sorted per-lane lists
#pragma unroll
  for (int j = 0; j < 32; ++j) { md[wv][lane * 32 + j] = bd[j]; mi[wv][lane * 32 + j] = bi[j]; }
  __syncthreads();

  // 32-way sorted merge: extract global min 32 times
  int hp = 0, keep = 0;
  for (int sel = 0; sel < 32; ++sel) {
    float hv = (hp < 32) ? md[wv][lane * 32 + hp] : 3.402823e38f;
    int   hi = (hp < 32) ? mi[wv][lane * 32 + hp] : 0x7fffffff;
    float bv = hv; int bix = hi; int bl = lane;
#pragma unroll
    for (int off = 16; off > 0; off >>= 1) {
      const float ov = __shfl_xor(bv, off);
      const int   oi = __shfl_xor(bix, off);
      const int   ol = __shfl_xor(bl, off);
      if (ov < bv || (ov == bv && (oi < bix || (oi == bix && ol < bl)))) { bv = ov; bix = oi; bl = ol; }
    }
    if (lane == sel) keep = bix;
    if (lane == bl) hp++;
  }
  idx_out[(size_t)col * K_ + lane] = keep;
}

// =====================================================================
// 2) Stats for the 8-channel weight-net BN (axes 0,2,3): one block per
//    channel, recompute s = wn_w . rel16 + b on the fly, LDS reduce.
// =====================================================================
__global__ __launch_bounds__(256) void wn_stats_kernel(const float* __restrict__ xyz,
                                                       const int* __restrict__ idx32,
                                                       const float* __restrict__ wn_w,
                                                       const float* __restrict__ wn_b,
                                                       float* __restrict__ acc_wn) {
  __shared__ float s1s[256], s2s[256];
  const int o = blockIdx.x;
  const float w0 = wn_w[o * 3], w1 = wn_w[o * 3 + 1], w2 = wn_w[o * 3 + 2], bb = wn_b[o];
  float s1 = 0.f, s2 = 0.f;
  for (int t = threadIdx.x; t < B_ * N_ * KPC_; t += 256) {
    const int b = t / (N_ * KPC_);
    const int r = t % (N_ * KPC_);
    const int n = r / KPC_;
    const int k = r % KPC_;
    const int i = idx32[((size_t)b * N_ + n) * K_ + k];
    const size_t xb = (size_t)b * 3 * N_;
    const float rx = xyz[xb + i] - xyz[xb + n];
    const float ry = xyz[xb + N_ + i] - xyz[xb + N_ + n];
    const float rz = xyz[xb + 2 * N_ + i] - xyz[xb + 2 * N_ + n];
    const float s = w0 * rx + w1 * ry + w2 * rz + bb;
    s1 += s; s2 += s * s;
  }
  s1s[threadIdx.x] = s1; s2s[threadIdx.x] = s2; __syncthreads();
  for (int off = 128; off > 0; off >>= 1) {
    if (threadIdx.x < off) { s1s[threadIdx.x] += s1s[threadIdx.x + off]; s2s[threadIdx.x] += s2s[threadIdx.x + off]; }
    __syncthreads();
  }
  if (threadIdx.x == 0) { acc_wn[o] = s1s[0]; acc_wn[WN_ + o] = s2s[0]; }
}

// =====================================================================
// 3) Fused point-conv: 16 points per block (4 waves x 4 points).
//    Phase 1 builds AGG (512 x 16) tile in LDS (weight-net + gathers),
//    Phase 2 runs PC = pc_lin_w(64x512) @ AGG as 4 WMMA tiles
//    (128 x v_wmma_f32_16x16x4_f32 per wave), B-operand from LDS.
// =====================================================================
__global__ __launch_bounds__(128) void pointconv_kernel(
    const float* __restrict__ xyz, const float* __restrict__ feats,
    const float* __restrict__ cost_volume, const int* __restrict__ idx32,
    const float* __restrict__ acc_wn,
    const float* __restrict__ wn_w, const float* __restrict__ wn_b,
    const float* __restrict__ wn_g, const float* __restrict__ wn_be,
    const float* __restrict__ lin_w, const float* __restrict__ lin_b,
    float* __restrict__ PC) {
  __shared__ float Xs[512 * 16];          // AGG tile (k-major, 16 cols)
  __shared__ float wL[4][WN_ * KPC_];
  __shared__ int   iL[4][KPC_];
  __shared__ float gL[4][C_ * KPC_];

  const int lane = threadIdx.x & 31;
  const int wv   = threadIdx.x >> 5;
  const int col0 = blockIdx.x * 16;

  for (int p = 0; p < 4; ++p) {
    const int pt  = wv * 4 + p;
    const int col = col0 + pt;
    const int b   = col / N_;
    const int n   = col % N_;
    if (lane < KPC_) {
      const int i = idx32[((size_t)b * N_ + n) * K_ + lane];
      iL[wv][lane] = i;
      const size_t xb = (size_t)b * 3 * N_;
      const float rx = xyz[xb + i] - xyz[xb + n];
      const float ry = xyz[xb + N_ + i] - xyz[xb + N_ + n];
      const float rz = xyz[xb + 2 * N_ + i] - xyz[xb + 2 * N_ + n];
#pragma unroll
      for (int o = 0; o < WN_; ++o) {
        const float s = wn_w[o * 3] * rx + wn_w[o * 3 + 1] * ry + wn_w[o * 3 + 2] * rz + wn_b[o];
        const float m = acc_wn[o] * (1.f / CNT_WN_);
        const float v = acc_wn[WN_ + o] * (1.f / CNT_WN_) - m * m;
        const float t = (s - m) * rsqrtf(v + EPS_) * wn_g[o] + wn_be[o];
        wL[wv][o * KPC_ + lane] = lrelu_(t);
      }
    }
    __syncthreads();
    for (int t = lane; t < C_ * KPC_; t += 32) {
      const int cch = t >> 4, k = t & 15;
      const int i = iL[wv][k];
      const size_t base = ((size_t)b * C_ + cch) * N_;
      gL[wv][t] = cost_volume[base + i] - feats[base + i];
    }
    __syncthreads();
    const int o = lane >> 2;
    const int cb = (lane & 3) * 16;
#pragma unroll
    for (int cc = 0; cc < 16; ++cc) {
      float acc = 0.f;
#pragma unroll
      for (int k = 0; k < KPC_; ++k)
        acc += wL[wv][o * KPC_ + k] * gL[wv][(cb + cc) * KPC_ + k];
      Xs[(o * C_ + cb + cc) * 16 + pt] = acc;    // f = o*64 + c
    }
    __syncthreads();
  }
  __syncthreads();

  // Phase 2: wave wv computes rows [wv*16, wv*16+16) of PC tile
  const int m0 = wv * 16;
  const int half = lane >> 4, l16 = lane & 15;
  v8f c = {};
  const float* Wr = lin_w + (size_t)(m0 + l16) * 512;
  for (int k0 = 0; k0 < 512; k0 += 4) {
    const v2f a = *(const v2f*)(Wr + k0 + 2 * half);
    v2f bm;
    bm.x = Xs[(k0 + 2 * half) * 16 + l16];
    bm.y = Xs[(k0 + 2 * half + 1) * 16 + l16];
    c = __builtin_amdgcn_wmma_f32_16x16x4_f32(false, a, false, bm, (short)0, c, false, false);
  }
  const int col = col0 + l16;
  const int b2 = col / N_, n2 = col % N_;
#pragma unroll
  for (int r = 0; r < 8; ++r) {
    const int m = m0 + r + 8 * half;
    PC[((size_t)b2 * C_ + m) * N_ + n2] = c[r] + lin_b[m];
  }
}

// =====================================================================
// 4) Generic f32 WMMA GEMM with bias: Y(B,M,N) = W(MxKD) @ X(B,KD,N)+b.
//    16-col slab per block; the X tile is staged global->LDS with the
//    CDNA5 async path (global_load_async_to_lds_b128 + s_wait_asynccnt),
//    then each wave computes a 16x16 tile with v_wmma_f32_16x16x4_f32.
// =====================================================================
template <int M, int KD>
__global__ __launch_bounds__(32 * (M / 16)) void gemm_bias(const float* __restrict__ W,
                                                           const float* __restrict__ bias,
                                                           const float* __restrict__ X,
                                                           float* __restrict__ Y) {
  constexpr int NT = 32 * (M / 16);
  __shared__ float Xs[KD * 16];
  const int lane = threadIdx.x & 31;
  const int wv   = threadIdx.x >> 5;
  const int col0 = blockIdx.x * 16;
  const int bcol  = col0 / N_;      // all 16 cols in one batch (N % 16 == 0)
  const int nbase = col0 % N_;

  // ---- async stage: each op moves 4 contiguous floats (16B aligned) ----
  const uint32_t ldsBase = (uint32_t)(uintptr_t)&Xs[0];
#pragma unroll
  for (int q = threadIdx.x; q < KD * 4; q += NT) {
    const int k  = q >> 2;
    const int p4 = q & 3;
    const float* gsrc = X + ((size_t)bcol * KD + k) * N_ + nbase + p4 * 4;
    const uint32_t laddr = ldsBase + (uint32_t)((k * 16 + p4 * 4) * 4);
    asm volatile("global_load_async_to_lds_b128 %0, %1, off"
                 :: "v"(laddr), "v"(gsrc) : "memory");
  }
  asm volatile("s_wait_asynccnt 0x0" ::: "memory");
  __syncthreads();

  const int m0 = wv * 16;
  const int half = lane >> 4, l16 = lane & 15;
  v8f c = {};
  const float* Wr = W + (size_t)(m0 + l16) * KD;
  for (int k0 = 0; k0 < KD; k0 += 4) {
    const v2f a = *(const v2f*)(Wr + k0 + 2 * half);
    v2f bm;
    bm.x = Xs[(k0 + 2 * half) * 16 + l16];
    bm.y = Xs[(k0 + 2 * half + 1) * 16 + l16];
    c = __builtin_amdgcn_wmma_f32_16x16x4_f32(false, a, false, bm, (short)0, c, false, false);
  }
  const int col = col0 + l16;
  const int b = col / N_, n = col % N_;
#pragma unroll
  for (int r = 0; r < 8; ++r) {
    const int m = m0 + r + 8 * half;
    Y[((size_t)b * M + m) * N_ + n] = c[r] + bias[m];
  }
}

// =====================================================================
// 5) Per-channel global stats (sum, sumsq over B*N): block per channel.
//    Channel count = gridDim.x.
// =====================================================================
__global__ __launch_bounds__(256) void stats_kernel(const float* __restrict__ Y,
                                                    float* __restrict__ acc) {
  __shared__ float s1s[256], s2s[256];
  const int c = blockIdx.x;
  const int C = gridDim.x;
  float s1 = 0.f, s2 = 0.f;
  for (int t = threadIdx.x; t < B_ * N_; t += 256) {
    const int b = t / N_, n = t % N_;
    const float v = Y[((size_t)b * C + c) * N_ + n];
    s1 += v; s2 += v * v;
  }
  s1s[threadIdx.x] = s1; s2s[threadIdx.x] = s2; __syncthreads();
  for (int off = 128; off > 0; off >>= 1) {
    if (threadIdx.x < off) { s1s[threadIdx.x] += s1s[threadIdx.x + off]; s2s[threadIdx.x] += s2s[threadIdx.x + off]; }
    __syncthreads();
  }
  if (threadIdx.x == 0) { acc[c] = s1s[0]; acc[C + c] = s2s[0]; }
}

// cost_mod = cost_volume + lrelu(bn(PC))
__global__ void pc_finalize_kernel(const float* __restrict__ PC, const float* __restrict__ cost_volume,
                                   const float* __restrict__ acc, const float* __restrict__ gamma,
                                   const float* __restrict__ beta, float* __restrict__ CM) {
  const int i = blockIdx.x * blockDim.x + threadIdx.x;
  if (i >= B_ * C_ * N_) return;
  const int c = (i / N_) % C_;
  const float inv = 1.f / CNT_BN_;
  const float m = acc[c] * inv;
  const float v = acc[C_ + c] * inv - m * m;
  const float t = (PC[i] - m) * rsqrtf(v + EPS_) * gamma[c] + beta[c];
  CM[i] = cost_volume[i] + lrelu_(t);
}

// in-place Y = lrelu(bn(Y)), C channels
__global__ void bn_lrelu_kernel(float* __restrict__ Y, const float* __restrict__ acc,
                                const float* __restrict__ gamma, const float* __restrict__ beta, int C) {
  const int i = blockIdx.x * blockDim.x + threadIdx.x;
  if (i >= B_ * C * N_) return;
  const int c = (i / N_) % C;
  const float inv = 1.f / CNT_BN_;
  const float m = acc[c] * inv;
  const float v = acc[C + c] * inv - m * m;
  const float t = (Y[i] - m) * rsqrtf(v + EPS_) * gamma[c] + beta[c];
  Y[i] = lrelu_(t);
}

// =====================================================================
// 6) Attention: one wave per point; lanes split D=128 (d = lane+32t).
//    scores via shuffle-reduced dots, softmax across lanes, output
//    X0 = tanh(sum_k attn_k * (VV+pos)).
// =====================================================================
__global__ __launch_bounds__(32) void attention_kernel(const float* __restrict__ Q,
                                                       const float* __restrict__ KK,
                                                       const float* __restrict__ VV,
                                                       const int* __restrict__ idx32,
                                                       const float* __restrict__ xyz,
                                                       const float* __restrict__ pos_w,
                                                       const float* __restrict__ pos_b,
                                                       float* __restrict__ X0) {
  const int lane = threadIdx.x;
  const int col = blockIdx.x;
  const int b = col / N_, n = col % N_;
  const size_t db = (size_t)b * D_ * N_;
  const size_t xb = (size_t)b * 3 * N_;

  float qt[4], pw0[4], pw1[4], pw2[4], pb[4];
#pragma unroll
  for (int t = 0; t < 4; ++t) {
    const int d = lane + 32 * t;
    qt[t] = Q[db + (size_t)d * N_ + n];
    pw0[t] = pos_w[d * 3]; pw1[t] = pos_w[d * 3 + 1]; pw2[t] = pos_w[d * 3 + 2];
    pb[t] = pos_b[d];
  }
  const float qx = xyz[xb + n], qy = xyz[xb + N_ + n], qz = xyz[xb + 2 * N_ + n];

  float myscore = 0.f;
  for (int k = 0; k < K_; ++k) {
    const int i = idx32[(size_t)col * K_ + k];
    const float rx = xyz[xb + i] - qx, ry = xyz[xb + N_ + i] - qy, rz = xyz[xb + 2 * N_ + i] - qz;
    float part = 0.f;
#pragma unroll
    for (int t = 0; t < 4; ++t) {
      const int d = lane + 32 * t;
      const float pos = pw0[t] * rx + pw1[t] * ry + pw2[t] * rz + pb[t];
      part += qt[t] * (KK[db + (size_t)d * N_ + i] + pos);
    }
#pragma unroll
    for (int off = 16; off > 0; off >>= 1) part += __shfl_xor(part, off);
    if (lane == k) myscore = part * 0.088388347648318447f;  // 1/sqrt(128)
  }
  float mx = myscore;
#pragma unroll
  for (int off = 16; off > 0; off >>= 1) mx = fmaxf(mx, __shfl_xor(mx, off));
  const float e = __expf(myscore - mx);
  float sum = e;
#pragma unroll
  for (int off = 16; off > 0; off >>= 1) sum += __shfl_xor(sum, off);
  const float attn = e / sum;

  float acc[4] = {0.f, 0.f, 0.f, 0.f};
  for (int k = 0; k < K_; ++k) {
    const float a = __shfl(attn, k);
    const int i = idx32[(size_t)col * K_ + k];
    const float rx = xyz[xb + i] - qx, ry = xyz[xb + N_ + i] - qy, rz = xyz[xb + 2 * N_ + i] - qz;
#pragma unroll
    for (int t = 0; t < 4; ++t) {
      const int d = lane + 32 * t;
      const float pos = pw0[t] * rx + pw1[t] * ry + pw2[t] * rz + pb[t];
      acc[t] += a * (VV[db + (size_t)d * N_ + i] + pos);
    }
  }
#pragma unroll
  for (int t = 0; t < 4; ++t) X0[db + (size_t)(lane + 32 * t) * N_ + n] = tanhf(acc[t]);
}

// flow = clip(fc_w @ x + fc_b, +-200)
__global__ void fc_kernel(const float* __restrict__ x, const float* __restrict__ fc_w,
                          const float* __restrict__ fc_b, float* __restrict__ flow) {
  const int i = blockIdx.x * blockDim.x + threadIdx.x;
  if (i >= B_ * 3 * N_) return;
  const int n = i % N_;
  const int o = (i / N_) % 3;
  const int b = i / (3 * N_);
  float s = fc_b[o];
  const float* wr = fc_w + o * C_;
  const float* xc = x + (size_t)b * C_ * N_ + n;
#pragma unroll 8
  for (int c = 0; c < C_; ++c) s += wr[c] * xc[(size_t)c * N_];
  flow[i] = fminf(200.f, fmaxf(-200.f, s));
}

// =====================================================================
extern "C" void kernel_launch(void* const* d_in, const int* in_sizes, int n_in,
                              void* d_out, int out_size, void* d_ws, size_t ws_size,
                              hipStream_t stream) {
  const float* xyz        = (const float*)d_in[0];
  const float* feats      = (const float*)d_in[1];
  const float* cost_vol   = (const float*)d_in[2];
  const float* pc_wn_w    = (const float*)d_in[3];
  const float* pc_wn_b    = (const float*)d_in[4];
  const float* pc_wn_g    = (const float*)d_in[5];
  const float* pc_wn_be   = (const float*)d_in[6];
  const float* pc_lin_w   = (const float*)d_in[7];
  const float* pc_lin_b   = (const float*)d_in[8];
  const float* pc_bn_g    = (const float*)d_in[9];
  const float* pc_bn_be   = (const float*)d_in[10];
  const float* q_w        = (const float*)d_in[11];
  const float* q_b        = (const float*)d_in[12];
  const float* k_w        = (const float*)d_in[13];
  const float* k_b        = (const float*)d_in[14];
  const float* v_w        = (const float*)d_in[15];
  const float* v_b        = (const float*)d_in[16];
  const float* pos_w      = (const float*)d_in[17];
  const float* pos_b      = (const float*)d_in[18];
  const float* mlp1_w     = (const float*)d_in[19];
  const float* mlp1_b     = (const float*)d_in[20];
  const float* mlp1_g     = (const float*)d_in[21];
  const float* mlp1_be    = (const float*)d_in[22];
  const float* mlp2_w     = (const float*)d_in[23];
  const float* mlp2_b     = (const float*)d_in[24];
  const float* mlp2_g     = (const float*)d_in[25];
  const float* mlp2_be    = (const float*)d_in[26];
  const float* fc_w       = (const float*)d_in[27];
  const float* fc_b       = (const float*)d_in[28];

  // ---- workspace layout (~26.5 MB of floats) ----
  float* ws = (float*)d_ws;
  int*   idx = (int*)d_ws;
  size_t off = (size_t)B_ * N_ * K_;         // idx32 ints
  float* acc_wn = ws + off; off += 2 * WN_;
  float* acc_pc = ws + off; off += 2 * C_;
  float* acc_m1 = ws + off; off += 2 * 128;
  float* acc_m2 = ws + off; off += 2 * 64;
  off = (off + 255) & ~(size_t)255;
  float* PC  = ws + off; off += (size_t)B_ * C_ * N_;
  float* CM  = ws + off; off += (size_t)B_ * C_ * N_;
  float* Qb  = ws + off; off += (size_t)B_ * D_ * N_;
  float* KKb = ws + off; off += (size_t)B_ * D_ * N_;
  float* VVb = ws + off; off += (size_t)B_ * D_ * N_;
  float* X0  = ws + off; off += (size_t)B_ * D_ * N_;
  float* H1  = ws + off; off += (size_t)B_ * D_ * N_;

  float* outX  = (float*)d_out;                 // x: (B, 64, N)
  float* flow  = outX + (size_t)B_ * C_ * N_;   // flow: (B, 3, N)

  const int colTiles = (B_ * N_) / 16;

  knn_kernel<<<(B_ * N_) / 4, 128, 0, stream>>>(xyz, idx);
  wn_stats_kernel<<<WN_, 256, 0, stream>>>(xyz, idx, pc_wn_w, pc_wn_b, acc_wn);
  pointconv_kernel<<<colTiles, 128, 0, stream>>>(xyz, feats, cost_vol, idx, acc_wn,
                                                 pc_wn_w, pc_wn_b, pc_wn_g, pc_wn_be,
                                                 pc_lin_w, pc_lin_b, PC);
  stats_kernel<<<C_, 256, 0, stream>>>(PC, acc_pc);
  pc_finalize_kernel<<<(B_ * C_ * N_ + 255) / 256, 256, 0, stream>>>(PC, cost_vol, acc_pc,
                                                                     pc_bn_g, pc_bn_be, CM);
  // Q/K/V projections commute with the neighbor gather: project once, gather later.
  gemm_bias<128, 64><<<colTiles, 256, 0, stream>>>(q_w, q_b, CM, Qb);
  gemm_bias<128, 64><<<colTiles, 256, 0, stream>>>(k_w, k_b, CM, KKb);
  gemm_bias<128, 64><<<colTiles, 256, 0, stream>>>(v_w, v_b, CM, VVb);
  attention_kernel<<<B_ * N_, 32, 0, stream>>>(Qb, KKb, VVb, idx, xyz, pos_w, pos_b, X0);
  gemm_bias<128, 128><<<colTiles, 256, 0, stream>>>(mlp1_w, mlp1_b, X0, H1);
  stats_kernel<<<128, 256, 0, stream>>>(H1, acc_m1);
  bn_lrelu_kernel<<<(B_ * 128 * N_ + 255) / 256, 256, 0, stream>>>(H1, acc_m1, mlp1_g, mlp1_be, 128);
  gemm_bias<64, 128><<<colTiles, 128, 0, stream>>>(mlp2_w, mlp2_b, H1, outX);
  stats_kernel<<<64, 256, 0, stream>>>(outX, acc_m2);
  bn_lrelu_kernel<<<(B_ * 64 * N_ + 255) / 256, 256, 0, stream>>>(outX, acc_m2, mlp2_g, mlp2_be, 64);
  fc_kernel<<<(B_ * 3 * N_ + 255) / 256, 256, 0, stream>>>(outX, fc_w, fc_b, flow);
}